// HitAtKLoss_49512382988771
// MI455X (gfx1250) — compile-verified
//
#include <hip/hip_runtime.h>
#include <math.h>

typedef float v2f __attribute__((ext_vector_type(2)));
typedef float v4f __attribute__((ext_vector_type(4)));
typedef float v8f __attribute__((ext_vector_type(8)));

#define B_SZ 128
#define D_SZ 1024
#define P_SZ 4
#define N_SZ 511
#define M_TOT (P_SZ + N_SZ)   /* 515 */
#define TEMP_INV 10.0f        /* 1/0.1 */
#define KTOP 3

// ---------------- kernel 0: anchor L2 norms (B values) ----------------
__global__ __launch_bounds__(256) void hitk_anchor_norm(const float* __restrict__ anchor,
                                                        float* __restrict__ anNorm) {
  __shared__ float sm[256];
  const int b = blockIdx.x, tid = threadIdx.x;
  const v4f* a = reinterpret_cast<const v4f*>(anchor + (size_t)b * D_SZ);
  v4f v = a[tid];                                  // 256 threads * 4 floats = 1024
  sm[tid] = v.x * v.x + v.y * v.y + v.z * v.z + v.w * v.w;
  __syncthreads();
  for (int off = 128; off > 0; off >>= 1) {
    if (tid < off) sm[tid] += sm[tid + off];
    __syncthreads();
  }
  if (tid == 0) anNorm[b] = sqrtf(sm[0]);
}

// ------- kernel 1: cosine sims via V_WMMA_F32_16X16X4_F32, fused norms -------
// One wave per (b, 16-row tile). A = 16 'others' rows, B = anchor broadcast
// across columns -> all 16 output columns identical, so lane L's accumulator
// element r = dot(anchor, row r + 8*(L>>4)).
//
// K-slot permutation trick: the dot product is order-invariant over K, so each
// lane loads one b128 float4: half-wave 0 covers global elems {k..k+3},
// half-wave 1 covers {k+4..k+7}. WMMA #1 consumes (x,y), WMMA #2 consumes
// (z,w), with the identical permutation applied to the anchor (B) fragment.
// This doubles per-request size vs the natural b64 fragment loads.
__global__ __launch_bounds__(32) void hitk_cossim_wmma(const float* __restrict__ anchor,
                                                       const float* __restrict__ others,
                                                       const float* __restrict__ anNorm,
                                                       float* __restrict__ simOut,
                                                       int M) {
  const int lane = threadIdx.x;
  const int half = lane >> 4;                 // half-wave id
  const int tile = blockIdx.x;
  const int b    = blockIdx.y;

  int m  = tile * 16 + (lane & 15);
  int mc = (m < M) ? m : (M - 1);             // clamp keeps loads in-bounds, branchless

  const float* __restrict__ row  = others + ((size_t)b * M + mc) * D_SZ + 4 * half;
  const float* __restrict__ arow = anchor + (size_t)b * D_SZ + 4 * half;

  v8f c = {0.f, 0.f, 0.f, 0.f, 0.f, 0.f, 0.f, 0.f};
  float sq = 0.f;
#pragma unroll 4
  for (int k = 0; k < D_SZ; k += 8) {
    // negatives stream: no reuse, 268 MB > L2 -> non-temporal b128
    v4f av4 = __builtin_nontemporal_load(reinterpret_cast<const v4f*>(row + k));
    // anchor: reused by all 33 tiles of this batch row -> regular temporal
    v4f bv4 = *reinterpret_cast<const v4f*>(arow + k);
    sq += av4.x * av4.x + av4.y * av4.y + av4.z * av4.z + av4.w * av4.w;
    v2f a0 = {av4.x, av4.y}, a1 = {av4.z, av4.w};
    v2f b0 = {bv4.x, bv4.y}, b1 = {bv4.z, bv4.w};
    c = __builtin_amdgcn_wmma_f32_16x16x4_f32(false, a0, false, b0, (short)0, c, false, false);
    c = __builtin_amdgcn_wmma_f32_16x16x4_f32(false, a1, false, b1, (short)0, c, false, false);
  }
  // complete per-row sum of squares: lanes L and L^16 cover complementary K halves
  sq += __shfl_xor(sq, 16, 32);
  const float an = anNorm[b];

#pragma unroll
  for (int r = 0; r < 8; ++r) {
    int   mrow = tile * 16 + half * 8 + r;
    float osq  = __shfl(sq, half * 8 + r, 32);          // norm^2 of row mrow
    float cs   = c[r] / fmaxf(an * sqrtf(osq), 1e-8f);
    if ((lane & 15) == 0 && mrow < M)                   // lane 0 -> rows 0..7, lane 16 -> 8..15
      simOut[(size_t)b * M + mrow] = cs;
  }
}

// ------- kernel 2: per-row softmax / top-3 / ranks -> row loss (one wave/row) -------
__global__ __launch_bounds__(32) void hitk_row_loss(const float* __restrict__ posSim,
                                                    const float* __restrict__ negSim,
                                                    float* __restrict__ rowLoss,
                                                    float* __restrict__ rowPosSum) {
  const int b = blockIdx.x, lane = threadIdx.x;
  const float NEG_INF = -__builtin_inff();
  float sv[17];                                          // ceil(515/32)
#pragma unroll
  for (int j = 0; j < 17; ++j) {
    int idx = lane + 32 * j;
    float v = NEG_INF;
    if (idx < M_TOT)
      v = (idx < P_SZ) ? posSim[b * P_SZ + idx]
                       : negSim[(size_t)b * N_SZ + (idx - P_SZ)];
    sv[j] = v;
  }
  // row max
  float mx = NEG_INF;
#pragma unroll
  for (int j = 0; j < 17; ++j) mx = fmaxf(mx, sv[j]);
  for (int off = 16; off > 0; off >>= 1) mx = fmaxf(mx, __shfl_xor(mx, off, 32));
  // softmax denominator (exp of -inf pads -> 0)
  float se = 0.f;
#pragma unroll
  for (int j = 0; j < 17; ++j) se += expf((sv[j] - mx) * TEMP_INV);
  for (int off = 16; off > 0; off >>= 1) se += __shfl_xor(se, off, 32);

  // top-3 (jax tie-break: first/smallest index) + positive prob mass in top-K
  int c0 = -1, c1 = -1, c2 = -1;
  float posTopK = 0.f;
  for (int t = 0; t < KTOP; ++t) {
    float bvv = NEG_INF; int bii = 0x7fffffff;
#pragma unroll
    for (int j = 0; j < 17; ++j) {
      int idx = lane + 32 * j;
      float v = sv[j];
      if (idx < M_TOT && idx != c0 && idx != c1 && idx != c2) {
        if (v > bvv || (v == bvv && idx < bii)) { bvv = v; bii = idx; }
      }
    }
    for (int off = 16; off > 0; off >>= 1) {
      float ov = __shfl_xor(bvv, off, 32);
      int   oi = __shfl_xor(bii, off, 32);
      if (ov > bvv || (ov == bvv && oi < bii)) { bvv = ov; bii = oi; }
    }
    if (t == 0) c0 = bii; else if (t == 1) c1 = bii; else c2 = bii;
    if (bii < P_SZ) posTopK += expf((bvv - mx) * TEMP_INV) / se;
  }

  // ranks of the 4 positives (stable-argsort semantics) + weighted prob
  float wsum = 0.f, psum = 0.f;
#pragma unroll
  for (int p = 0; p < P_SZ; ++p) {
    float sp = posSim[b * P_SZ + p];
    psum += sp;
    int cnt = 0;
#pragma unroll
    for (int j = 0; j < 17; ++j) {
      int idx = lane + 32 * j;
      float v = sv[j];
      if (idx < M_TOT && (v > sp || (v == sp && idx < p))) cnt++;
    }
    for (int off = 16; off > 0; off >>= 1) cnt += __shfl_xor(cnt, off, 32);
    float rank = (float)cnt + 1.0f;
    float w = 1.0f / (rank + 1.0f);
    if (rank <= (float)KTOP) w *= (float)(KTOP + 1);
    wsum += (expf((sp - mx) * TEMP_INV) / se) * w;
  }

  if (lane == 0) {
    float hit = 0.7f * posTopK + 0.3f * wsum;
    rowLoss[b]   = -logf(hit + 1e-8f);
    rowPosSum[b] = psum;
  }
}

// ---------------- kernel 3: final reduction over B rows ----------------
__global__ __launch_bounds__(128) void hitk_finalize(const float* __restrict__ rowLoss,
                                                     const float* __restrict__ rowPosSum,
                                                     float* __restrict__ out) {
  __shared__ float s1[128], s2[128];
  const int t = threadIdx.x;
  s1[t] = rowLoss[t];
  s2[t] = rowPosSum[t];
  __syncthreads();
  for (int off = 64; off > 0; off >>= 1) {
    if (t < off) { s1[t] += s1[t + off]; s2[t] += s2[t + off]; }
    __syncthreads();
  }
  if (t == 0)
    out[0] = s1[0] / (float)B_SZ +
             0.5f * (1.0f - s2[0] / (float)(B_SZ * P_SZ));
}

extern "C" void kernel_launch(void* const* d_in, const int* in_sizes, int n_in,
                              void* d_out, int out_size, void* d_ws, size_t ws_size,
                              hipStream_t stream) {
  const float* anchor    = (const float*)d_in[0];
  const float* positives = (const float*)d_in[1];
  const float* negatives = (const float*)d_in[2];
  float* out = (float*)d_out;

  float* ws        = (float*)d_ws;
  float* anNorm    = ws;                         // 128
  float* posSim    = anNorm + B_SZ;              // 128*4
  float* negSim    = posSim + B_SZ * P_SZ;       // 128*511
  float* rowLoss   = negSim + (size_t)B_SZ * N_SZ;
  float* rowPosSum = rowLoss + B_SZ;             // total ~66.3K floats (~259 KB)

  hitk_anchor_norm<<<B_SZ, 256, 0, stream>>>(anchor, anNorm);

  dim3 gpos(1, B_SZ);
  dim3 gneg((N_SZ + 15) / 16, B_SZ);
  hitk_cossim_wmma<<<gpos, 32, 0, stream>>>(anchor, positives, anNorm, posSim, P_SZ);
  hitk_cossim_wmma<<<gneg, 32, 0, stream>>>(anchor, negatives, anNorm, negSim, N_SZ);

  hitk_row_loss<<<B_SZ, 32, 0, stream>>>(posSim, negSim, rowLoss, rowPosSum);
  hitk_finalize<<<1, 128, 0, stream>>>(rowLoss, rowPosSum, out);
}